// MIPTVDepthEstimator_25477746000507
// MI455X (gfx1250) — compile-verified
//
#include <hip/hip_runtime.h>
#include <stdint.h>

// Problem geometry (fixed by the reference).
#define B_   16
#define H_   512
#define W_   512
#define HW_  (H_ * W_)
#define BHW_ ((size_t)B_ * (size_t)HW_)
#define KRAD 15            // (31-1)/2

#define ALPHA_  0.15f
#define MU_     10.0f
#define TAU_    0.125f
#define SIG_    0.125f
#define EPS_    1e-6f
#define INV1PTAU (1.0f / (1.0f + 0.125f))

struct QState { unsigned prefix; unsigned rem; };

// ---------------------------------------------------------------------------
// CDNA5 async global->LDS helpers (ASYNCcnt path, CDNA5 ISA).
// Flat shared-memory addresses map to LDS via their low 32 bits (ISA §10.2).
// ---------------------------------------------------------------------------
__device__ __forceinline__ uint32_t lds_addr32(const void* p) {
  return (uint32_t)(uintptr_t)p;
}

__device__ __forceinline__ void async_load_b128_to_lds(void* lds_dst, const float* gsrc) {
#if defined(__gfx1250__)
  asm volatile("global_load_async_to_lds_b128 %0, %1, off"
               :
               : "v"(lds_addr32(lds_dst)), "v"((uint64_t)(uintptr_t)gsrc)
               : "memory");
#else
  ((volatile float*)lds_dst)[0] = gsrc[0];
  ((volatile float*)lds_dst)[1] = gsrc[1];
  ((volatile float*)lds_dst)[2] = gsrc[2];
  ((volatile float*)lds_dst)[3] = gsrc[3];
#endif
}

__device__ __forceinline__ void wait_async0() {
#if defined(__gfx1250__)
#if __has_builtin(__builtin_amdgcn_s_wait_asynccnt)
  __builtin_amdgcn_s_wait_asynccnt(0);
#else
  asm volatile("s_wait_asynccnt 0" ::: "memory");
#endif
#endif
}

__device__ __forceinline__ float clampf(float v, float lo, float hi) {
  return fminf(fmaxf(v, lo), hi);
}

__device__ __forceinline__ void ld4(const float* __restrict__ p, float o[4]) {
  const float4 v = *(const float4*)p;
  o[0] = v.x; o[1] = v.y; o[2] = v.z; o[3] = v.w;
}
__device__ __forceinline__ void st4(float* __restrict__ p, const float v[4]) {
  float4 t; t.x = v[0]; t.y = v[1]; t.z = v[2]; t.w = v[3];
  *(float4*)p = t;
}

// ---------------------------------------------------------------------------
// 1) Horizontal 31-tap max per (b, c, y) row. Row staged in LDS via one
//    global_load_async_to_lds_b128 per lane (128 lanes fill 512 floats).
// ---------------------------------------------------------------------------
__global__ void hmax31_kernel(const float* __restrict__ in, float* __restrict__ out) {
  __shared__ float srow[W_];
  const int row = blockIdx.x;                 // 0 .. B*3*H-1
  const float* src = in + (size_t)row * W_;
  float* dst = out + (size_t)row * W_;
  const int t = threadIdx.x;                  // 256 threads, 2 output px each

  if (t < 128) async_load_b128_to_lds(&srow[t * 4], src + t * 4);
  wait_async0();
  __syncthreads();

#pragma unroll
  for (int i = 0; i < 2; ++i) {
    const int x  = t + i * 256;
    int x0 = x - KRAD; if (x0 < 0) x0 = 0;
    int x1 = x + KRAD; if (x1 > W_ - 1) x1 = W_ - 1;
    float m = srow[x0];
    for (int xx = x0 + 1; xx <= x1; ++xx) m = fmaxf(m, srow[xx]);
    dst[x] = m;
  }
}

// ---------------------------------------------------------------------------
// 2) Vertical 31-tap max on all 3 channels + combine: S = |Rmax-max(Gmax,Bmax)|+R
//    hmax output (50 MB) is L2-resident on MI455X (192 MB L2); loads coalesce in x.
// ---------------------------------------------------------------------------
__global__ void vmax_combine_kernel(const float* __restrict__ hm,
                                    const float* __restrict__ Iy,
                                    float* __restrict__ S) {
  const size_t idx = (size_t)blockIdx.x * blockDim.x + threadIdx.x;
  if (idx >= BHW_) return;
  const int b = (int)(idx / HW_);
  const int r = (int)(idx % HW_);
  const int y = r / W_, x = r % W_;
  int y0 = y - KRAD; if (y0 < 0) y0 = 0;
  int y1 = y + KRAD; if (y1 > H_ - 1) y1 = H_ - 1;
  const float* rp = hm + (size_t)(b * 3 + 0) * HW_;
  const float* gp = hm + (size_t)(b * 3 + 1) * HW_;
  const float* bp = hm + (size_t)(b * 3 + 2) * HW_;
  float rm = -3.402823466e38f, gm = rm, bm = rm;
  for (int yy = y0; yy <= y1; ++yy) {
    const size_t o = (size_t)yy * W_ + x;
    rm = fmaxf(rm, rp[o]); gm = fmaxf(gm, gp[o]); bm = fmaxf(bm, bp[o]);
  }
  const float Rv = Iy[(size_t)(b * 3) * HW_ + r];
  S[idx] = fabsf(rm - fmaxf(gm, bm)) + Rv;
}

// ---------------------------------------------------------------------------
// Exact order-statistic selection via 3-pass radix on float bit patterns.
// All selected arrays are >= 0, so uint bit order == value order.
// Pass 1: bits[31:20] (4096 bins, LDS-privatized histogram per block).
// Pass 2: bits[19:8]  (4096 bins, prefix-filtered, global atomics).
// Pass 3: bits[7:0]   (256 bins,  prefix-filtered, global atomics).
// ---------------------------------------------------------------------------
__global__ void qhist1_kernel(const float* __restrict__ src, unsigned* __restrict__ hist) {
  __shared__ unsigned lh[4096];
  const int b = blockIdx.x >> 6;      // 64 chunks per batch
  const int c = blockIdx.x & 63;
  for (int i = threadIdx.x; i < 4096; i += 256) lh[i] = 0u;
  __syncthreads();
  const float* p = src + (size_t)b * HW_ + (size_t)c * 4096;
#pragma unroll
  for (int k = 0; k < 16; ++k) {
    const unsigned bits = __float_as_uint(p[threadIdx.x + k * 256]);
    atomicAdd(&lh[bits >> 20], 1u);
  }
  __syncthreads();
  for (int i = threadIdx.x; i < 4096; i += 256)
    if (lh[i]) atomicAdd(&hist[(size_t)b * 4096 + i], lh[i]);
}

__global__ void qscan1_kernel(const unsigned* __restrict__ hist, QState* __restrict__ st,
                              int P, int r0, int r1, int r2, int r3) {
  const int bp = blockIdx.x;
  if (threadIdx.x != 0) return;
  const int b = bp / P, p = bp % P;
  const int ranks[4] = {r0, r1, r2, r3};
  const unsigned rank = (unsigned)ranks[p];
  const unsigned* h = hist + (size_t)b * 4096;
  unsigned cum = 0;
  for (int i = 0; i < 4096; ++i) {
    const unsigned c = h[i];
    if (cum + c > rank) {
      st[bp].prefix = (unsigned)i << 20;
      st[bp].rem    = rank - cum;
      return;
    }
    cum += c;
  }
  st[bp].prefix = 0xFFFu << 20; st[bp].rem = 0;   // unreachable for valid ranks
}

__global__ void qhist23_kernel(const float* __restrict__ src, const QState* __restrict__ st,
                               unsigned* __restrict__ hist, int P,
                               unsigned mask, int shift, int nbins) {
  const int b = blockIdx.x >> 6;
  const int c = blockIdx.x & 63;
  unsigned pref[4];
  for (int p = 0; p < P; ++p) pref[p] = st[b * P + p].prefix;
  const float* sp = src + (size_t)b * HW_ + (size_t)c * 4096;
#pragma unroll
  for (int k = 0; k < 16; ++k) {
    const unsigned bits   = __float_as_uint(sp[threadIdx.x + k * 256]);
    const unsigned masked = bits & mask;
    const unsigned bin    = (bits >> shift) & (unsigned)(nbins - 1);
    for (int p = 0; p < P; ++p)
      if (masked == pref[p])
        atomicAdd(&hist[(size_t)(b * P + p) * nbins + bin], 1u);
  }
}

__global__ void qscan23_kernel(const unsigned* __restrict__ hist, QState* __restrict__ st,
                               int P, int shift, int nbins, int is_final,
                               float* __restrict__ results) {
  const int bp = blockIdx.x;
  if (threadIdx.x != 0) return;
  const unsigned* h = hist + (size_t)bp * nbins;
  const unsigned rank = st[bp].rem;
  unsigned cum = 0;
  for (int i = 0; i < nbins; ++i) {
    const unsigned c = h[i];
    if (cum + c > rank) {
      const unsigned pfx = st[bp].prefix | ((unsigned)i << shift);
      st[bp].prefix = pfx;
      st[bp].rem    = rank - cum;
      if (is_final) results[bp] = __uint_as_float(pfx);
      return;
    }
    cum += c;
  }
}

__global__ void finalize_s_kernel(const float* __restrict__ results, float* __restrict__ qp,
                                  float fracLo, float fracHi) {
  const int b = threadIdx.x;
  if (b >= B_) return;
  const float v0 = results[b * 4 + 0], v1 = results[b * 4 + 1];
  const float v2 = results[b * 4 + 2], v3 = results[b * 4 + 3];
  const float lo = v0 + (v1 - v0) * fracLo;
  const float hi = v2 + (v3 - v2) * fracHi;
  qp[b * 4 + 0] = lo;
  qp[b * 4 + 1] = 1.0f / (hi - lo + EPS_);
}

__global__ void finalize_g_kernel(const float* __restrict__ results, float* __restrict__ qp,
                                  float fracMed) {
  const int b = threadIdx.x;
  if (b >= B_) return;
  const float v0 = results[b * 2 + 0], v1 = results[b * 2 + 1];
  float s = v0 + (v1 - v0) * fracMed;
  s = fmaxf(s, EPS_);
  qp[b * 4 + 2] = 1.0f / s;
}

// ---------------------------------------------------------------------------
// 3) N_hat = clip((S - lo) * scale, 0, 1)
// ---------------------------------------------------------------------------
__global__ void nhat_kernel(const float* __restrict__ S, const float* __restrict__ qp,
                            float* __restrict__ Nh) {
  const size_t idx = (size_t)blockIdx.x * blockDim.x + threadIdx.x;
  if (idx >= BHW_) return;
  const int b = (int)(idx / HW_);
  const float v = (S[idx] - qp[b * 4 + 0]) * qp[b * 4 + 1];
  Nh[idx] = fminf(fmaxf(v, 0.0f), 1.0f);
}

// 4) grad magnitude of N_hat (forward differences, zero-padded)
__global__ void grad_kernel(const float* __restrict__ Nh, float* __restrict__ G) {
  const size_t idx = (size_t)blockIdx.x * blockDim.x + threadIdx.x;
  if (idx >= BHW_) return;
  const int r = (int)(idx % HW_);
  const int y = r / W_, x = r % W_;
  const float c = Nh[idx];
  const float dx = (x < W_ - 1) ? Nh[idx + 1]  - c : 0.0f;
  const float dy = (y < H_ - 1) ? Nh[idx + W_] - c : 0.0f;
  G[idx] = sqrtf(dx * dx + dy * dy + EPS_);
}

// 5) Wx/Wy weights + PD state init (u = ubar = N_hat, px = py = 0)
__global__ void prep_kernel(const float* __restrict__ Nh, const float* __restrict__ qp,
                            float* __restrict__ Wxp, float* __restrict__ Wyp,
                            float* __restrict__ u, float* __restrict__ ub,
                            float* __restrict__ px, float* __restrict__ py) {
  const size_t idx = (size_t)blockIdx.x * blockDim.x + threadIdx.x;
  if (idx >= BHW_) return;
  const int b = (int)(idx / HW_);
  const int r = (int)(idx % HW_);
  const int y = r / W_, x = r % W_;
  const float c  = Nh[idx];
  const float dx = (x < W_ - 1) ? Nh[idx + 1]  - c : 0.0f;
  const float dy = (y < H_ - 1) ? Nh[idx + W_] - c : 0.0f;
  const float isg = qp[b * 4 + 2];
  Wxp[idx] = 1.0f + MU_ * expf(-fabsf(dx) * isg);
  Wyp[idx] = 1.0f + MU_ * expf(-fabsf(dy) * isg);
  u[idx] = c; ub[idx] = c; px[idx] = 0.0f; py[idx] = 0.0f;
}

// ---------------------------------------------------------------------------
// 6) One fused primal-dual step, 4-wide along x (W divisible by 4).
//    All bulk traffic is b128; x-neighbor dual values come from registers and
//    same-iteration neighbor px/py are RECOMPUTED from ubar, so one launch
//    covers dual + primal updates. Working set (~168 MB) is L2-resident on
//    MI455X (192 MB L2) -> minimizing vmem instruction count is what matters.
// ---------------------------------------------------------------------------
__global__ void pd_step4_kernel(const float* __restrict__ ub_in,
                                const float* __restrict__ px_in,
                                const float* __restrict__ py_in,
                                float* __restrict__ ub_out,
                                float* __restrict__ px_out,
                                float* __restrict__ py_out,
                                float* __restrict__ u,
                                const float* __restrict__ Nh,
                                const float* __restrict__ Wxp,
                                const float* __restrict__ Wyp,
                                float* __restrict__ outp, int write_final) {
  const size_t q = (size_t)blockIdx.x * blockDim.x + threadIdx.x;   // quad id
  if (q >= BHW_ / 4) return;
  const size_t i = q * 4;
  const int r = (int)(i % HW_);
  const int y = r / W_, x = r % W_;          // x is a multiple of 4

  float ub[4], pxv[4], pyv[4], wx[4], wy[4], uo[4], nh[4];
  ld4(ub_in + i, ub);  ld4(px_in + i, pxv); ld4(py_in + i, pyv);
  ld4(Wxp  + i, wx);   ld4(Wyp  + i, wy);
  ld4(u    + i, uo);   ld4(Nh   + i, nh);

  const bool has_r = (x + 4 < W_);
  const bool has_l = (x > 0);
  const bool has_d = (y < H_ - 1);
  const bool has_u = (y > 0);

  const float ubr = has_r ? ub_in[i + 4] : 0.0f;
  const float ubl = has_l ? ub_in[i - 1] : 0.0f;

  float ubd[4] = {0.f, 0.f, 0.f, 0.f};
  float ubu[4] = {0.f, 0.f, 0.f, 0.f};
  if (has_d) ld4(ub_in + i + W_, ubd);
  if (has_u) ld4(ub_in + i - W_, ubu);

  float pxn[4], pyn[4], dv[4];
#pragma unroll
  for (int j = 0; j < 4; ++j) {
    const float dx = (j < 3) ? (ub[j + 1] - ub[j]) : (has_r ? ubr - ub[3] : 0.0f);
    const float dy = has_d ? (ubd[j] - ub[j]) : 0.0f;
    const float bx = wx[j] * ALPHA_;
    const float by = wy[j] * ALPHA_;
    pxn[j] = clampf(pxv[j] + SIG_ * dx, -bx, bx);
    pyn[j] = clampf(pyv[j] + SIG_ * dy, -by, by);
  }

  float pxl_new = 0.0f;                      // same-iteration left-neighbor px
  if (has_l) {
    const float bxl = Wxp[i - 1] * ALPHA_;
    pxl_new = clampf(px_in[i - 1] + SIG_ * (ub[0] - ubl), -bxl, bxl);
  }
  dv[0] = pxn[0] - pxl_new;
  dv[1] = pxn[1] - pxn[0];
  dv[2] = pxn[2] - pxn[1];
  dv[3] = pxn[3] - pxn[2];

  if (has_u) {                               // same-iteration upper-neighbor py
    float pyu[4], wyu[4];
    ld4(py_in + i - W_, pyu);
    ld4(Wyp   + i - W_, wyu);
#pragma unroll
    for (int j = 0; j < 4; ++j) {
      const float byu = wyu[j] * ALPHA_;
      const float pyn_u = clampf(pyu[j] + SIG_ * (ub[j] - ubu[j]), -byu, byu);
      dv[j] += pyn[j] - pyn_u;
    }
  } else {
#pragma unroll
    for (int j = 0; j < 4; ++j) dv[j] += pyn[j];
  }

  float un[4], ubn[4], fo[4];
#pragma unroll
  for (int j = 0; j < 4; ++j) {
    un[j]  = (uo[j] + TAU_ * (dv[j] + nh[j])) * INV1PTAU;
    ubn[j] = 2.0f * un[j] - uo[j];            // theta = 1
    fo[j]  = clampf(un[j], 0.0f, 1.0f);
  }
  st4(u + i, un);
  st4(ub_out + i, ubn);
  st4(px_out + i, pxn);
  st4(py_out + i, pyn);
  if (write_final) st4(outp + i, fo);
}

// ---------------------------------------------------------------------------
// Host-side orchestration
// ---------------------------------------------------------------------------
static void run_select(const float* src, QState* st, unsigned* hist, float* results,
                       int P, int r0, int r1, int r2, int r3, hipStream_t stream) {
  hipMemsetAsync(hist, 0, (size_t)B_ * 4096 * sizeof(unsigned), stream);
  qhist1_kernel<<<B_ * 64, 256, 0, stream>>>(src, hist);
  qscan1_kernel<<<B_ * P, 32, 0, stream>>>(hist, st, P, r0, r1, r2, r3);

  hipMemsetAsync(hist, 0, (size_t)B_ * P * 4096 * sizeof(unsigned), stream);
  qhist23_kernel<<<B_ * 64, 256, 0, stream>>>(src, st, hist, P, 0xFFF00000u, 8, 4096);
  qscan23_kernel<<<B_ * P, 32, 0, stream>>>(hist, st, P, 8, 4096, 0, results);

  hipMemsetAsync(hist, 0, (size_t)B_ * P * 256 * sizeof(unsigned), stream);
  qhist23_kernel<<<B_ * 64, 256, 0, stream>>>(src, st, hist, P, 0xFFFFFF00u, 0, 256);
  qscan23_kernel<<<B_ * P, 32, 0, stream>>>(hist, st, P, 0, 256, 1, results);
}

extern "C" void kernel_launch(void* const* d_in, const int* in_sizes, int n_in,
                              void* d_out, int out_size, void* d_ws, size_t ws_size,
                              hipStream_t stream) {
  (void)in_sizes; (void)n_in; (void)out_size; (void)ws_size;
  const float* Iy  = (const float*)d_in[0];
  float* outp = (float*)d_out;

  // Workspace layout (floats), with phase-based reuse:
  float* f    = (float*)d_ws;
  float* tmp3 = f;                 // [3*BHW] hmax out  -> later px_a / py_a / u
  float* S    = f + 3 * BHW_;      // [BHW]  S          -> later ubar_a
  float* Nh   = f + 4 * BHW_;      // [BHW]  persistent
  float* Wxp  = f + 5 * BHW_;      // [BHW]  persistent
  float* Wyp  = f + 6 * BHW_;      // [BHW]  persistent
  float* G    = f + 7 * BHW_;      // [BHW]  grad mag   -> later ubar_b
  float* px_b = f + 8 * BHW_;      // [BHW]
  float* py_b = f + 9 * BHW_;      // [BHW]
  float* px_a = tmp3;
  float* py_a = tmp3 + BHW_;
  float* u    = tmp3 + 2 * BHW_;
  float* ub_a = S;
  float* ub_b = G;
  unsigned* hist   = (unsigned*)(f + 10 * BHW_);        // up to B*4*4096 u32 = 1 MB
  QState*  st      = (QState*)(hist + (size_t)B_ * 4 * 4096);
  float* results   = (float*)(st + B_ * 4);
  float* qparams   = results + B_ * 4;

  const int nblk  = (int)((BHW_ + 255) / 256);
  const int nblk4 = (int)((BHW_ / 4 + 255) / 256);

  // Stage 1-2: separable 31x31 max pool + S = |Rmax - max(Gmax,Bmax)| + R
  hmax31_kernel<<<B_ * 3 * H_, 256, 0, stream>>>(Iy, tmp3);
  vmax_combine_kernel<<<nblk, 256, 0, stream>>>(tmp3, Iy, S);

  // Quantile positions (match jnp.quantile 'linear')
  const double posLo = 0.01 * (HW_ - 1), posHi = 0.99 * (HW_ - 1), posMed = 0.5 * (HW_ - 1);
  const int iLo = (int)posLo, iHi = (int)posHi, iMed = (int)posMed;
  const float fLo  = (float)(posLo  - iLo);
  const float fHi  = (float)(posHi  - iHi);
  const float fMed = (float)(posMed - iMed);

  // Stage 3: exact p01/p99 of S per batch -> lo, 1/(hi-lo+eps)
  run_select(S, st, hist, results, 4, iLo, iLo + 1, iHi, iHi + 1, stream);
  finalize_s_kernel<<<1, 32, 0, stream>>>(results, qparams, fLo, fHi);
  nhat_kernel<<<nblk, 256, 0, stream>>>(S, qparams, Nh);

  // Stage 4: exact median of |grad N_hat| per batch -> 1/sigma
  grad_kernel<<<nblk, 256, 0, stream>>>(Nh, G);
  run_select(G, st, hist, results, 2, iMed, iMed + 1, 0, 0, stream);
  finalize_g_kernel<<<1, 32, 0, stream>>>(results, qparams, fMed);

  // Stage 5: weights + PD init (G buffer is free after this point -> ub_b)
  prep_kernel<<<nblk, 256, 0, stream>>>(Nh, qparams, Wxp, Wyp, u, ub_a, px_a, py_a);

  // Stage 6: 30 fused primal-dual iterations, ping-ponging A/B buffer sets.
  for (int t = 0; t < 30; ++t) {
    const bool even = (t & 1) == 0;
    const float* ubi = even ? ub_a : ub_b;
    const float* pxi = even ? px_a : px_b;
    const float* pyi = even ? py_a : py_b;
    float* ubo = even ? ub_b : ub_a;
    float* pxo = even ? px_b : px_a;
    float* pyo = even ? py_b : py_a;
    const int final_iter = (t == 29) ? 1 : 0;
    pd_step4_kernel<<<nblk4, 256, 0, stream>>>(ubi, pxi, pyi, ubo, pxo, pyo,
                                               u, Nh, Wxp, Wyp, outp, final_iter);
  }
}